// GraphMedNCA_37142877176008
// MI455X (gfx1250) — compile-verified
//
#include <hip/hip_runtime.h>
#include <hip/hip_bf16.h>
#include <math.h>

typedef __attribute__((ext_vector_type(2))) float v2f;
typedef __attribute__((ext_vector_type(4))) float v4f;
typedef __attribute__((ext_vector_type(8))) float v8f;

#define BATCH 8
#define HW    64
#define NNODE 4096          // 64*64
#define CFEAT 16
#define KNN   8
#define HEADS 4
#define DH    4

// ---------------------------------------------------------------------------
// Kernel 1: conv1 (1->64, 3x3, pad 1) + BN + ReLU
// out: h64 [B][64][4096]
// ---------------------------------------------------------------------------
__global__ __launch_bounds__(256) void k_conv1(
    const float* __restrict__ x, const float* __restrict__ w,
    const float* __restrict__ bias,
    const float* __restrict__ g, const float* __restrict__ bb,
    const float* __restrict__ m, const float* __restrict__ v,
    float* __restrict__ out)
{
    int tid = blockIdx.x * blockDim.x + threadIdx.x;   // b*64*4096 + oc*4096 + p
    if (tid >= BATCH * 64 * NNODE) return;
    int p  = tid & (NNODE - 1);
    int oc = (tid >> 12) & 63;
    int b  = tid >> 18;
    int yy = p >> 6, xx = p & 63;
    const float* xb = x + (size_t)b * NNODE;
    const float* wk = w + oc * 9;
    float acc = 0.0f;
#pragma unroll
    for (int ky = 0; ky < 3; ++ky) {
        int iy = yy + ky - 1;
        if (iy < 0 || iy >= HW) continue;
#pragma unroll
        for (int kx = 0; kx < 3; ++kx) {
            int ix = xx + kx - 1;
            if (ix < 0 || ix >= HW) continue;
            acc += xb[iy * HW + ix] * wk[ky * 3 + kx];
        }
    }
    float s   = g[oc] * rsqrtf(v[oc] + 1e-5f);
    float val = (acc + bias[oc] - m[oc]) * s + bb[oc];
    out[tid]  = fmaxf(val, 0.0f);
}

// ---------------------------------------------------------------------------
// Kernel 2: conv2 (64->16, 3x3, pad 1) + BN + ReLU
// out: h16 [B][16][4096] (channel-major, for residual) and feat [B][4096][16]
// ---------------------------------------------------------------------------
__global__ __launch_bounds__(256) void k_conv2(
    const float* __restrict__ h64, const float* __restrict__ w,
    const float* __restrict__ bias,
    const float* __restrict__ g, const float* __restrict__ bb,
    const float* __restrict__ m, const float* __restrict__ v,
    float* __restrict__ h16, float* __restrict__ feat)
{
    int tid = blockIdx.x * blockDim.x + threadIdx.x;   // b*16*4096 + oc*4096 + p
    if (tid >= BATCH * CFEAT * NNODE) return;
    int p  = tid & (NNODE - 1);
    int oc = (tid >> 12) & 15;
    int b  = tid >> 16;
    int yy = p >> 6, xx = p & 63;
    float acc = 0.0f;
    for (int ic = 0; ic < 64; ++ic) {
        const float* cb = h64 + (((size_t)b * 64 + ic) << 12);
        const float* wk = w + ((size_t)oc * 64 + ic) * 9;
#pragma unroll
        for (int ky = 0; ky < 3; ++ky) {
            int iy = yy + ky - 1;
            if (iy < 0 || iy >= HW) continue;
#pragma unroll
            for (int kx = 0; kx < 3; ++kx) {
                int ix = xx + kx - 1;
                if (ix < 0 || ix >= HW) continue;
                acc += cb[iy * HW + ix] * wk[ky * 3 + kx];
            }
        }
    }
    float s   = g[oc] * rsqrtf(v[oc] + 1e-5f);
    float val = (acc + bias[oc] - m[oc]) * s + bb[oc];
    val = fmaxf(val, 0.0f);
    h16[tid] = val;
    feat[(((size_t)b * NNODE) + p) * CFEAT + oc] = val;
}

// ---------------------------------------------------------------------------
// Kernel 3: per node: squared norm, Wx = nodes * gat_w^T, attention scalars
// ---------------------------------------------------------------------------
__global__ __launch_bounds__(256) void k_prep(
    const float* __restrict__ feat, const float* __restrict__ gat_w,
    const float* __restrict__ att_src, const float* __restrict__ att_dst,
    float* __restrict__ sq, float* __restrict__ wx,
    float* __restrict__ asrc, float* __restrict__ adst)
{
    int tid = blockIdx.x * blockDim.x + threadIdx.x;   // b*4096 + n
    if (tid >= BATCH * NNODE) return;
    const float* f = feat + (size_t)tid * CFEAT;
    float fr[CFEAT];
    float s2 = 0.0f;
#pragma unroll
    for (int c = 0; c < CFEAT; ++c) { fr[c] = f[c]; s2 += fr[c] * fr[c]; }
    sq[tid] = s2;
    float w[CFEAT];
#pragma unroll
    for (int o = 0; o < CFEAT; ++o) {
        float a = 0.0f;
#pragma unroll
        for (int c = 0; c < CFEAT; ++c) a += gat_w[o * CFEAT + c] * fr[c];
        w[o] = a;
        wx[(size_t)tid * CFEAT + o] = a;
    }
#pragma unroll
    for (int h = 0; h < HEADS; ++h) {
        float as = 0.0f, ad = 0.0f;
#pragma unroll
        for (int d = 0; d < DH; ++d) {
            as += att_src[h * DH + d] * w[h * DH + d];
            ad += att_dst[h * DH + d] * w[h * DH + d];
        }
        asrc[(size_t)tid * HEADS + h] = as;
        adst[(size_t)tid * HEADS + h] = ad;
    }
}

// ---------------------------------------------------------------------------
// Kernel 4: kNN via fp32 WMMA.  8 waves per block, one 16-query tile per wave.
// dot tile D = A(16 col-nodes x 16 feats) * B(16 feats x 16 queries),
// built from 4 chained V_WMMA_F32_16X16X4_F32.  Rank by sq[m] - 2*dot
// (sq[q] is constant per row -> rank-invariant).
// ---------------------------------------------------------------------------
__global__ __launch_bounds__(256) void k_knn(
    const float* __restrict__ feat, const float* __restrict__ sq,
    int* __restrict__ knn)
{
    const int b    = blockIdx.y;
    const int wave = threadIdx.x >> 5;
    const int qt   = blockIdx.x * 8 + wave;   // query tile owned by this wave
    const int lane = threadIdx.x & 31;
    const int half = lane >> 4;               // 0 or 1
    const int l15  = lane & 15;
    const int q    = qt * 16 + l15;           // query tracked by this lane

    const float* fb  = feat + (size_t)b * NNODE * CFEAT;
    const float* sqb = sq + (size_t)b * NNODE;

    // B operand (query tile), loop invariant.  Layout (4x16 fp32):
    // v0: lanes<16 -> K=4kc+0, lanes>=16 -> K=4kc+2;  v1: +1 / +3.  N = lane&15.
    v2f bq[4];
#pragma unroll
    for (int kc = 0; kc < 4; ++kc) {
        const float* p = fb + (size_t)q * CFEAT + kc * 4 + half * 2;
        v2f t; t.x = p[0]; t.y = p[1];
        bq[kc] = t;
    }

    float d[KNN]; int id[KNN];
#pragma unroll
    for (int i = 0; i < KNN; ++i) { d[i] = __builtin_inff(); id[i] = -1; }

    for (int mt = 0; mt < NNODE / 16; ++mt) {
        const int m0 = mt * 16;
        if (mt + 1 < NNODE / 16)
            __builtin_prefetch(fb + (size_t)(m0 + 16 + l15) * CFEAT, 0, 1);

        v8f c = {0.f, 0.f, 0.f, 0.f, 0.f, 0.f, 0.f, 0.f};
#pragma unroll
        for (int kc = 0; kc < 4; ++kc) {
            const float* pa = fb + (size_t)(m0 + l15) * CFEAT + kc * 4 + half * 2;
            v2f a; a.x = pa[0]; a.y = pa[1];
            // (neg_a, A, neg_b, B, c_mod, C, reuse_a, reuse_b)
            c = __builtin_amdgcn_wmma_f32_16x16x4_f32(
                    false, a, false, bq[kc], (short)0, c, false, false);
        }

        // sq for the 8 candidate rows of this lane: two aligned float4 loads
        const float* psq = sqb + m0 + 8 * half;
        v4f sq_lo = *(const v4f*)(psq);
        v4f sq_hi = *(const v4f*)(psq + 4);

        // D layout: vgpr r, lanes<16 -> M=r, lanes>=16 -> M=8+r; N = lane&15.
        // Compute all candidates before divergent insertion.
        float s[KNN];
#pragma unroll
        for (int r = 0; r < 4; ++r) {
            s[r]     = sq_lo[r] - 2.0f * c[r];
            s[r + 4] = sq_hi[r] - 2.0f * c[r + 4];
        }
#pragma unroll
        for (int r = 0; r < KNN; ++r) {
            int mg = m0 + 8 * half + r;
            if (mg != q && s[r] < d[7]) {
                d[7] = s[r]; id[7] = mg;
#pragma unroll
                for (int j = 7; j > 0; --j) {
                    if (d[j] < d[j - 1]) {
                        float td = d[j]; d[j] = d[j - 1]; d[j - 1] = td;
                        int   ti = id[j]; id[j] = id[j - 1]; id[j - 1] = ti;
                    }
                }
            }
        }
    }

    // merge the two lane halves (both track query q)
#pragma unroll
    for (int j = 0; j < KNN; ++j) {
        float od = __shfl_xor(d[j], 16, 32);
        int   oi = __shfl_xor(id[j], 16, 32);
        if (od < d[7]) {
            d[7] = od; id[7] = oi;
#pragma unroll
            for (int t = 7; t > 0; --t) {
                if (d[t] < d[t - 1]) {
                    float td = d[t]; d[t] = d[t - 1]; d[t - 1] = td;
                    int   ti = id[t]; id[t] = id[t - 1]; id[t - 1] = ti;
                }
            }
        }
    }

    if (half == 0) {
        int* o = knn + ((size_t)b * NNODE + q) * KNN;
#pragma unroll
        for (int j = 0; j < KNN; ++j) o[j] = id[j];
    }
}

// ---------------------------------------------------------------------------
// Kernel 5: GAT gather + softmax + aggregate + bias + ReLU -> p [B][N][16]
// ---------------------------------------------------------------------------
__global__ __launch_bounds__(256) void k_gat(
    const float* __restrict__ wx, const float* __restrict__ asrc,
    const float* __restrict__ adst, const int* __restrict__ knn,
    const float* __restrict__ gat_bias, float* __restrict__ pout)
{
    int tid = blockIdx.x * blockDim.x + threadIdx.x;   // b*4096 + n
    if (tid >= BATCH * NNODE) return;
    int b = tid >> 12;
    size_t bN = (size_t)b * NNODE;

    int nbr[KNN + 1];
    const int* kn = knn + (size_t)tid * KNN;
#pragma unroll
    for (int k = 0; k < KNN; ++k) nbr[k] = kn[k];
    nbr[KNN] = tid & (NNODE - 1);                      // self loop

    float ad[HEADS];
#pragma unroll
    for (int h = 0; h < HEADS; ++h) ad[h] = adst[(size_t)tid * HEADS + h];

    float e[KNN + 1][HEADS];
    float mx[HEADS];
#pragma unroll
    for (int h = 0; h < HEADS; ++h) mx[h] = -__builtin_inff();
    for (int k = 0; k <= KNN; ++k) {
        const float* as = asrc + (bN + nbr[k]) * HEADS;
#pragma unroll
        for (int h = 0; h < HEADS; ++h) {
            float z = as[h] + ad[h];
            z = (z > 0.0f) ? z : 0.2f * z;             // leaky relu
            e[k][h] = z;
            mx[h] = fmaxf(mx[h], z);
        }
    }
    float ssum[HEADS] = {0.f, 0.f, 0.f, 0.f};
    for (int k = 0; k <= KNN; ++k)
#pragma unroll
        for (int h = 0; h < HEADS; ++h) {
            e[k][h] = __expf(e[k][h] - mx[h]);
            ssum[h] += e[k][h];
        }
    float inv[HEADS];
#pragma unroll
    for (int h = 0; h < HEADS; ++h) inv[h] = 1.0f / ssum[h];

    float out[CFEAT];
#pragma unroll
    for (int c = 0; c < CFEAT; ++c) out[c] = 0.0f;
    for (int k = 0; k <= KNN; ++k) {
        const float* wn = wx + (bN + nbr[k]) * CFEAT;
#pragma unroll
        for (int h = 0; h < HEADS; ++h) {
            float a = e[k][h] * inv[h];
#pragma unroll
            for (int dd = 0; dd < DH; ++dd)
                out[h * DH + dd] += a * wn[h * DH + dd];
        }
    }
    float* po = pout + (size_t)tid * CFEAT;
#pragma unroll
    for (int c = 0; c < CFEAT; ++c)
        po[c] = fmaxf(out[c] + gat_bias[c], 0.0f);
}

// ---------------------------------------------------------------------------
// threefry2x32 (JAX default PRNG), key(42) -> (0, 42)
// ---------------------------------------------------------------------------
__device__ inline void threefry2x32(unsigned k0, unsigned k1,
                                    unsigned c0, unsigned c1,
                                    unsigned& r0, unsigned& r1)
{
    const unsigned ks2 = k0 ^ k1 ^ 0x1BD11BDAu;
    unsigned ks[3] = {k0, k1, ks2};
    const int rot[8] = {13, 15, 26, 6, 17, 29, 16, 24};
    unsigned x0 = c0 + k0, x1 = c1 + k1;
#pragma unroll
    for (int j = 0; j < 5; ++j) {
#pragma unroll
        for (int i = 0; i < 4; ++i) {
            x0 += x1;
            int r = rot[(j & 1) * 4 + i];
            x1 = (x1 << r) | (x1 >> (32 - r));
            x1 ^= x0;
        }
        x0 += ks[(j + 1) % 3];
        x1 += ks[(j + 2) % 3] + (unsigned)(j + 1);
    }
    r0 = x0; r1 = x1;
}

// ---------------------------------------------------------------------------
// Kernel 6: update MLP (16->128->16) + stochastic mask + residual +
//           out projection + sigmoid -> d_out [B][1][64][64]
// ---------------------------------------------------------------------------
__global__ __launch_bounds__(256) void k_update(
    const float* __restrict__ pfeat, const float* __restrict__ h16,
    const float* __restrict__ w1, const float* __restrict__ b1,
    const float* __restrict__ w2, const float* __restrict__ b2,
    const float* __restrict__ out_w, const float* __restrict__ out_b,
    float* __restrict__ out)
{
    __shared__ float w1s[128 * 16];
    __shared__ float w2s[16 * 128];   // transposed: [c][o]
    __shared__ float b1s[128], b2s[16], ows[16];

    for (int i = threadIdx.x; i < 128 * 16; i += blockDim.x) w1s[i] = w1[i];
    for (int i = threadIdx.x; i < 16 * 128; i += blockDim.x) {
        int o = i >> 4, c = i & 15;                    // w2 row-major [o][c]
        w2s[c * 128 + o] = w2[i];
    }
    for (int i = threadIdx.x; i < 128; i += blockDim.x) b1s[i] = b1[i];
    if (threadIdx.x < 16) { b2s[threadIdx.x] = b2[threadIdx.x];
                            ows[threadIdx.x] = out_w[threadIdx.x]; }
    __syncthreads();

    int tid = blockIdx.x * blockDim.x + threadIdx.x;   // b*4096 + n
    if (tid >= BATCH * NNODE) return;
    int b = tid >> 12, n = tid & (NNODE - 1);

    float pv[CFEAT];
    const float* pf = pfeat + (size_t)tid * CFEAT;
#pragma unroll
    for (int c = 0; c < CFEAT; ++c) pv[c] = pf[c];

    float u2[CFEAT];
#pragma unroll
    for (int c = 0; c < CFEAT; ++c) u2[c] = b2s[c];
    for (int o = 0; o < 128; ++o) {
        float a = b1s[o];
        const float* wr = &w1s[o * 16];
#pragma unroll
        for (int c = 0; c < CFEAT; ++c) a += wr[c] * pv[c];
        a = fmaxf(a, 0.0f);
#pragma unroll
        for (int c = 0; c < CFEAT; ++c) u2[c] += w2s[c * 128 + o] * a;
    }

    // fire mask = (uniform(key(42), 32768) < 0.5)
    const unsigned HALF = (BATCH * NNODE) / 2;
    unsigned i = (unsigned)tid, r0, r1, bits;
    if (i < HALF) { threefry2x32(0u, 42u, i, i + HALF, r0, r1); bits = r0; }
    else          { threefry2x32(0u, 42u, i - HALF, i, r0, r1); bits = r1; }
    unsigned fb = (bits >> 9) | 0x3F800000u;
    float u = __uint_as_float(fb) - 1.0f;
    float mask = (u < 0.5f) ? 1.0f : 0.0f;

    float z = out_b[0];
#pragma unroll
    for (int c = 0; c < CFEAT; ++c) {
        float hn = h16[(((size_t)b * CFEAT + c) << 12) + n] + mask * u2[c];
        z += ows[c] * hn;
    }
    out[tid] = 1.0f / (1.0f + __expf(-z));
}

// ---------------------------------------------------------------------------
extern "C" void kernel_launch(void* const* d_in, const int* in_sizes, int n_in,
                              void* d_out, int out_size, void* d_ws, size_t ws_size,
                              hipStream_t stream) {
    const float* x       = (const float*)d_in[0];
    const float* enc_w1  = (const float*)d_in[1];
    const float* enc_b1  = (const float*)d_in[2];
    const float* bn1_g   = (const float*)d_in[3];
    const float* bn1_b   = (const float*)d_in[4];
    const float* bn1_m   = (const float*)d_in[5];
    const float* bn1_v   = (const float*)d_in[6];
    const float* enc_w2  = (const float*)d_in[7];
    const float* enc_b2  = (const float*)d_in[8];
    const float* bn2_g   = (const float*)d_in[9];
    const float* bn2_b   = (const float*)d_in[10];
    const float* bn2_m   = (const float*)d_in[11];
    const float* bn2_v   = (const float*)d_in[12];
    const float* gat_w   = (const float*)d_in[13];
    const float* att_src = (const float*)d_in[14];
    const float* att_dst = (const float*)d_in[15];
    const float* gat_bias= (const float*)d_in[16];
    const float* upd_w1  = (const float*)d_in[17];
    const float* upd_b1  = (const float*)d_in[18];
    const float* upd_w2  = (const float*)d_in[19];
    const float* upd_b2  = (const float*)d_in[20];
    const float* out_w   = (const float*)d_in[21];
    const float* out_b   = (const float*)d_in[22];
    float* out = (float*)d_out;

    // workspace carve (floats)
    float* ws   = (float*)d_ws;
    float* h64  = ws;                                  // 8*64*4096
    float* h16  = h64  + (size_t)BATCH * 64 * NNODE;   // 8*16*4096
    float* feat = h16  + (size_t)BATCH * CFEAT * NNODE;// 8*4096*16
    float* sq   = feat + (size_t)BATCH * NNODE * CFEAT;// 8*4096
    float* wx   = sq   + (size_t)BATCH * NNODE;        // 8*4096*16
    float* asrc = wx   + (size_t)BATCH * NNODE * CFEAT;// 8*4096*4
    float* adst = asrc + (size_t)BATCH * NNODE * HEADS;// 8*4096*4
    int*   knn  = (int*)(adst + (size_t)BATCH * NNODE * HEADS); // 8*4096*8
    float* pfeat= (float*)(knn + (size_t)BATCH * NNODE * KNN);  // 8*4096*16

    k_conv1<<<(BATCH * 64 * NNODE + 255) / 256, 256, 0, stream>>>(
        x, enc_w1, enc_b1, bn1_g, bn1_b, bn1_m, bn1_v, h64);

    k_conv2<<<(BATCH * CFEAT * NNODE + 255) / 256, 256, 0, stream>>>(
        h64, enc_w2, enc_b2, bn2_g, bn2_b, bn2_m, bn2_v, h16, feat);

    k_prep<<<(BATCH * NNODE + 255) / 256, 256, 0, stream>>>(
        feat, gat_w, att_src, att_dst, sq, wx, asrc, adst);

    dim3 kgrid(NNODE / 16 / 8, BATCH);     // 8 query-tiles (waves) per block
    k_knn<<<kgrid, 256, 0, stream>>>(feat, sq, knn);

    k_gat<<<(BATCH * NNODE + 255) / 256, 256, 0, stream>>>(
        wx, asrc, adst, knn, gat_bias, pfeat);

    k_update<<<(BATCH * NNODE + 255) / 256, 256, 0, stream>>>(
        pfeat, h16, upd_w1, upd_b1, upd_w2, upd_b2, out_w, out_b, out);
}